// GNN_7_52243982188637
// MI455X (gfx1250) — compile-verified
//
#include <hip/hip_runtime.h>
#include <hip/hip_bf16.h>

typedef float v2f __attribute__((ext_vector_type(2)));
typedef float v8f __attribute__((ext_vector_type(8)));

#define BM 64
#define BN 32
#define KC 32
#define LDA 36   // padded LDS row stride (multiple of 4 -> 16B-aligned B128 writes)

// ---------------------------------------------------------------------------
// gfx1250 async global->LDS copy (ASYNCcnt-tracked, no VGPR round trip).
// LDS address is taken from the vdst VGPR; generic LDS pointer keeps offset in [31:0].
// ---------------------------------------------------------------------------
__device__ __forceinline__ void async_copy_b128(void* lds_ptr, const float* gptr)
{
    unsigned lds = (unsigned)(uintptr_t)lds_ptr;
    unsigned long long ga = (unsigned long long)(uintptr_t)gptr;
    asm volatile("global_load_async_to_lds_b128 %0, %1, off"
                 :: "v"(lds), "v"(ga) : "memory");
}
__device__ __forceinline__ void async_wait0()
{
    asm volatile("s_wait_asynccnt 0" ::: "memory");
}

// ---------------------------------------------------------------------------
// Fused dual-GEMM + bias + ReLU:
//   Out[M,N] = act( A[M,K] @ W1[K,N]  (+ X[M,K] @ W2[K,N] if hasRoot) + bias )
// f32 WMMA 16x16x4; 8 waves/block -> 64x32 output tile.
// Double-buffered LDS with async-to-LDS pipeline: issue chunk k+1, compute
// chunk k, s_wait_asynccnt 0, barrier. Ragged tiles use a guarded scalar path.
// ---------------------------------------------------------------------------
__global__ __launch_bounds__(256) void gemm_dual_relu(
    const float* __restrict__ A, const float* __restrict__ X,
    const float* __restrict__ W1, const float* __restrict__ W2,
    const float* __restrict__ bias, float* __restrict__ Out,
    int M, int K, int N, int hasRoot, int doRelu)
{
    __shared__ float As[2][BM][LDA];
    __shared__ float Xs[2][BM][LDA];
    __shared__ float B1s[2][KC][LDA];
    __shared__ float B2s[2][KC][LDA];

    const int tid  = threadIdx.x;
    const int lane = tid & 31;
    const int wave = tid >> 5;      // 0..7
    const int wm   = wave & 3;      // M sub-tile 0..3
    const int wn   = wave >> 2;     // N sub-tile 0..1
    const int bm0  = blockIdx.x * BM;
    const int bn0  = blockIdx.y * BN;

    const int r    = lane & 15;             // M (A) / N (B) index within tile
    const int koff = (lane >> 4) << 1;      // K offset: 0 (lanes 0-15) or 2

    const bool fullM = (bm0 + BM <= M);

    auto load_chunk = [&](int kc, int buf) {
        if (fullM && (kc + KC <= K)) {
            // Async DMA path: one B128 per thread per tile slice.
            int mr = tid >> 3;            // 0..31
            int c4 = (tid & 7) << 2;      // 0,4,...,28
            const float* gA = A + (size_t)(bm0 + mr) * K + kc + c4;
            async_copy_b128(&As[buf][mr][c4],      gA);
            async_copy_b128(&As[buf][mr + 32][c4], gA + (size_t)32 * K);
            const float* gB = W1 + (size_t)(kc + mr) * N + bn0 + c4;
            async_copy_b128(&B1s[buf][mr][c4], gB);
            if (hasRoot) {
                const float* gX = X + (size_t)(bm0 + mr) * K + kc + c4;
                async_copy_b128(&Xs[buf][mr][c4],      gX);
                async_copy_b128(&Xs[buf][mr + 32][c4], gX + (size_t)32 * K);
                const float* gB2 = W2 + (size_t)(kc + mr) * N + bn0 + c4;
                async_copy_b128(&B2s[buf][mr][c4], gB2);
            }
        } else {
            // Guarded scalar path (K=5 layer, ragged last M block): zero-pads.
            for (int i = tid; i < BM * KC; i += 256) {
                int m = i >> 5, kk = i & 31;
                int gm = bm0 + m, gk = kc + kk;
                float av = 0.f, xv = 0.f;
                if (gm < M && gk < K) {
                    av = A[(size_t)gm * K + gk];
                    if (hasRoot) xv = X[(size_t)gm * K + gk];
                }
                As[buf][m][kk] = av;
                Xs[buf][m][kk] = xv;
            }
            for (int i = tid; i < KC * BN; i += 256) {
                int kk = i >> 5, n = i & 31;
                int gk = kc + kk;
                float b1 = 0.f, b2 = 0.f;
                if (gk < K) {
                    b1 = W1[(size_t)gk * N + bn0 + n];
                    if (hasRoot) b2 = W2[(size_t)gk * N + bn0 + n];
                }
                B1s[buf][kk][n] = b1;
                B2s[buf][kk][n] = b2;
            }
        }
    };

    v8f acc = {0.f, 0.f, 0.f, 0.f, 0.f, 0.f, 0.f, 0.f};

    load_chunk(0, 0);
    async_wait0();
    __syncthreads();

    int buf = 0;
    for (int kc = 0; kc < K; kc += KC) {
        int kn = kc + KC;
        if (kn < K) load_chunk(kn, buf ^ 1);   // DMA overlaps the WMMAs below

        if (hasRoot) {
#pragma unroll
            for (int k4 = 0; k4 < KC; k4 += 4) {
                v2f a, b;
                a.x = As[buf][16 * wm + r][k4 + koff];
                a.y = As[buf][16 * wm + r][k4 + koff + 1];
                b.x = B1s[buf][k4 + koff][16 * wn + r];
                b.y = B1s[buf][k4 + koff + 1][16 * wn + r];
                acc = __builtin_amdgcn_wmma_f32_16x16x4_f32(
                        false, a, false, b, (short)0, acc, false, false);
                v2f ax, b2;
                ax.x = Xs[buf][16 * wm + r][k4 + koff];
                ax.y = Xs[buf][16 * wm + r][k4 + koff + 1];
                b2.x = B2s[buf][k4 + koff][16 * wn + r];
                b2.y = B2s[buf][k4 + koff + 1][16 * wn + r];
                acc = __builtin_amdgcn_wmma_f32_16x16x4_f32(
                        false, ax, false, b2, (short)0, acc, false, false);
            }
        } else {
#pragma unroll
            for (int k4 = 0; k4 < KC; k4 += 4) {
                v2f a, b;
                a.x = As[buf][16 * wm + r][k4 + koff];
                a.y = As[buf][16 * wm + r][k4 + koff + 1];
                b.x = B1s[buf][k4 + koff][16 * wn + r];
                b.y = B1s[buf][k4 + koff + 1][16 * wn + r];
                acc = __builtin_amdgcn_wmma_f32_16x16x4_f32(
                        false, a, false, b, (short)0, acc, false, false);
            }
        }

        async_wait0();          // chunk k+1 DMA done (no-op on fallback path)
        __syncthreads();
        buf ^= 1;
    }

    // Epilogue: D layout — VGPR i: lanes 0-15 -> M=i, N=lane; lanes 16-31 -> M=i+8
    int   ncol  = bn0 + 16 * wn + r;
    float bv    = bias ? bias[ncol] : 0.f;
    int   mbase = bm0 + 16 * wm + ((lane >> 4) << 3);
#pragma unroll
    for (int i = 0; i < 8; i++) {
        int m = mbase + i;
        if (m < M) {
            float v = acc[i] + bv;
            if (doRelu) v = fmaxf(v, 0.f);
            Out[(size_t)m * N + ncol] = v;
        }
    }
}

// ---------------------------------------------------------------------------
// Edge scatter: agg[dst[e], c] += x[src[e], c] * w[e]
// agg (<=102 MB) is L2-resident on MI455X (192 MB L2) -> atomics stay on-die.
// ---------------------------------------------------------------------------
__global__ void scatter_add1(const float* __restrict__ x, const int* __restrict__ src,
                             const int* __restrict__ dst, const float* __restrict__ ew,
                             float* __restrict__ agg, int nEdges, int C)
{
    long long idx = (long long)blockIdx.x * blockDim.x + threadIdx.x;
    long long total = (long long)nEdges * C;
    if (idx >= total) return;
    int e = (int)(idx / C);
    int c = (int)(idx % C);
    int s = src[e], d = dst[e];
    atomicAdd(&agg[(size_t)d * C + c], x[(size_t)s * C + c] * ew[e]);
}

__global__ void scatter_add4(const float* __restrict__ x, const int* __restrict__ src,
                             const int* __restrict__ dst, const float* __restrict__ ew,
                             float* __restrict__ agg, int nEdges, int C)
{
    int cq4 = C >> 2;
    long long idx = (long long)blockIdx.x * blockDim.x + threadIdx.x;
    long long total = (long long)nEdges * cq4;
    if (idx >= total) return;
    int e  = (int)(idx / cq4);
    int cq = (int)(idx % cq4);
    int s = src[e], d = dst[e];
    float w = ew[e];
    float4 xv = ((const float4*)(x + (size_t)s * C))[cq];
    float* p  = agg + (size_t)d * C + 4 * cq;
    atomicAdd(p + 0, xv.x * w);
    atomicAdd(p + 1, xv.y * w);
    atomicAdd(p + 2, xv.z * w);
    atomicAdd(p + 3, xv.w * w);
}

// ---------------------------------------------------------------------------
// Pooling
// ---------------------------------------------------------------------------
__global__ void count_nodes(const int* __restrict__ batch, float* __restrict__ counts, int n)
{
    int i = blockIdx.x * blockDim.x + threadIdx.x;
    if (i < n) atomicAdd(&counts[batch[i]], 1.0f);
}

__global__ void pool_sum(const float* __restrict__ x, const int* __restrict__ batch,
                         float* __restrict__ pooled, int n, int C)
{
    long long idx = (long long)blockIdx.x * blockDim.x + threadIdx.x;
    long long total = (long long)n * C;
    if (idx >= total) return;
    int i = (int)(idx / C);
    int c = (int)(idx % C);
    atomicAdd(&pooled[(size_t)batch[i] * C + c], x[(size_t)i * C + c]);
}

__global__ void pool_div(float* __restrict__ pooled, const float* __restrict__ counts,
                         int G, int C)
{
    int idx = blockIdx.x * blockDim.x + threadIdx.x;
    if (idx >= G * C) return;
    int g = idx / C;
    pooled[idx] /= fmaxf(counts[g], 1.0f);
}

// ---------------------------------------------------------------------------
// Final 64 -> 1 projection (per-graph dot product)
// ---------------------------------------------------------------------------
__global__ void final_linear(const float* __restrict__ h, const float* __restrict__ W,
                             const float* __restrict__ b, float* __restrict__ out,
                             int G, int C)
{
    int g = blockIdx.x * blockDim.x + threadIdx.x;
    if (g >= G) return;
    float s = b[0];
    for (int c = 0; c < C; ++c) s += h[(size_t)g * C + c] * W[c];
    out[g] = s;
}

// ---------------------------------------------------------------------------
extern "C" void kernel_launch(void* const* d_in, const int* in_sizes, int n_in,
                              void* d_out, int out_size, void* d_ws, size_t ws_size,
                              hipStream_t stream)
{
    (void)in_sizes; (void)n_in; (void)out_size; (void)ws_size;
    const int NN = 50000, NE = 600000, NG = 256;
    static const int CH[8] = {5, 32, 128, 256, 512, 512, 256, 256};

    const float* x0    = (const float*)d_in[0];
    const int*   ei    = (const int*)d_in[1];
    const float* ew    = (const float*)d_in[2];
    const int*   batch = (const int*)d_in[3];
    const int*   src   = ei;
    const int*   dst   = ei + NE;

    // Workspace carve-up (floats)
    float* base   = (float*)d_ws;
    size_t big    = (size_t)NN * 512;
    float* bufA   = base;
    float* bufB   = bufA + big;
    float* agg    = bufB + big;
    float* pooled = agg + big;
    float* counts = pooled + (size_t)NG * 512;
    float* h1     = counts + NG;
    float* h2     = h1 + (size_t)NG * 256;

    const float* xcur = x0;
    float* xnext = bufA;

    for (int l = 0; l < 7; ++l) {
        int cin = CH[l], cout = CH[l + 1];
        const float* Wrel  = (const float*)d_in[4 + 3 * l + 0];
        const float* brel  = (const float*)d_in[4 + 3 * l + 1];
        const float* Wroot = (const float*)d_in[4 + 3 * l + 2];

        hipMemsetAsync(agg, 0, (size_t)NN * cin * sizeof(float), stream);

        if ((cin & 3) == 0) {
            long long t4 = (long long)NE * (cin >> 2);
            scatter_add4<<<(unsigned)((t4 + 255) / 256), 256, 0, stream>>>(
                xcur, src, dst, ew, agg, NE, cin);
        } else {
            long long t1 = (long long)NE * cin;
            scatter_add1<<<(unsigned)((t1 + 255) / 256), 256, 0, stream>>>(
                xcur, src, dst, ew, agg, NE, cin);
        }

        dim3 g((NN + BM - 1) / BM, cout / BN);
        gemm_dual_relu<<<g, 256, 0, stream>>>(agg, xcur, Wrel, Wroot, brel, xnext,
                                              NN, cin, cout, 1, 1);
        xcur  = xnext;
        xnext = (xnext == bufA) ? bufB : bufA;
    }

    // Mean pooling over graphs (final features: 256 channels)
    hipMemsetAsync(counts, 0, NG * sizeof(float), stream);
    hipMemsetAsync(pooled, 0, (size_t)NG * 256 * sizeof(float), stream);
    count_nodes<<<(NN + 255) / 256, 256, 0, stream>>>(batch, counts, NN);
    long long pt = (long long)NN * 256;
    pool_sum<<<(unsigned)((pt + 255) / 256), 256, 0, stream>>>(xcur, batch, pooled, NN, 256);
    pool_div<<<(NG * 256 + 255) / 256, 256, 0, stream>>>(pooled, counts, NG, 256);

    // MLP head (reuses the WMMA GEMM, root path disabled)
    const float* mW1 = (const float*)d_in[25]; const float* mb1 = (const float*)d_in[26];
    const float* mW2 = (const float*)d_in[27]; const float* mb2 = (const float*)d_in[28];
    const float* mW3 = (const float*)d_in[29]; const float* mb3 = (const float*)d_in[30];
    const float* oW  = (const float*)d_in[31]; const float* ob  = (const float*)d_in[32];

    gemm_dual_relu<<<dim3((NG + BM - 1) / BM, 256 / BN), 256, 0, stream>>>(
        pooled, nullptr, mW1, nullptr, mb1, h1, NG, 256, 256, 0, 1);
    gemm_dual_relu<<<dim3((NG + BM - 1) / BM, 128 / BN), 256, 0, stream>>>(
        h1, nullptr, mW2, nullptr, mb2, h2, NG, 256, 128, 0, 1);
    gemm_dual_relu<<<dim3((NG + BM - 1) / BM, 64 / BN), 256, 0, stream>>>(
        h2, nullptr, mW3, nullptr, mb3, h1, NG, 128, 64, 0, 1);

    final_linear<<<1, 256, 0, stream>>>(h1, oW, ob, (float*)d_out, NG, 64);
}